// Caps_Layer_24292335026824
// MI455X (gfx1250) — compile-verified
//
#include <hip/hip_runtime.h>
#include <hip/hip_bf16.h>

typedef __attribute__((ext_vector_type(2))) float v2f;
typedef __attribute__((ext_vector_type(8))) float v8f;
typedef __attribute__((ext_vector_type(4))) int   v4i;
typedef __attribute__((address_space(1))) v4i*    gv4i_p;  // global int4*
typedef __attribute__((address_space(3))) v4i*    lv4i_p;  // LDS int4*

#define NUM_CAP 5
#define DIM_CAP 5
#define NCOL    25          // NUM_CAP * DIM_CAP
#define SEQ     512
#define DIN     120
#define ROUTS   4
#define CHUNK   128         // rows of x staged per LDS pass
#define NCHUNK  (SEQ / CHUNK)
#define KSTEPS  (DIN / 4)   // 30 fp32-WMMA K steps
#define XPITCH  124         // x-chunk LDS row pitch (floats); stride%64=60 -> 4-way max
#define WPITCH  132         // W^T LDS col pitch; stride%64=4 -> conflict-free over 16 lanes
#define UPITCH  33          // u_hat LDS row pitch (>=32 so both N-tiles store unguarded; odd -> conflict-free)

#define SMEM_FLOATS (CHUNK*XPITCH + 32*WPITCH + SEQ*UPITCH + 8*NCOL + NCOL + 8)

__device__ __forceinline__ void wait_async_zero() {
#if __has_builtin(__builtin_amdgcn_s_wait_asynccnt)
    __builtin_amdgcn_s_wait_asynccnt(0);
#else
    asm volatile("s_wait_asynccnt 0x0" ::: "memory");
#endif
}

__global__ __launch_bounds__(256)
void caps_fused_kernel(const float* __restrict__ x,
                       const float* __restrict__ W,
                       float* __restrict__ out)
{
    extern __shared__ float smem[];
    float* xch  = smem;                      // [CHUNK][XPITCH]
    float* wT   = xch  + CHUNK * XPITCH;     // [32][WPITCH]  (W transposed, zero-padded)
    float* uh   = wT   + 32 * WPITCH;        // [SEQ][UPITCH] u_hat for this batch
    float* sred = uh   + SEQ * UPITCH;       // [8 waves][25]
    float* sS   = sred + 8 * NCOL;           // [25] reduced s
    float* sInv = sS   + NCOL;               // [5]  1/||s_i||

    const int tid  = threadIdx.x;
    const int lane = tid & 31;
    const int wid  = tid >> 5;
    const int b    = blockIdx.x;
    const float* xb = x + (size_t)b * SEQ * DIN;

    // ---- stage W^T into LDS, zero-padded to 32 cols x 128 k ----
    for (int i = tid; i < 32 * 128; i += 256) {
        int c = i >> 7, k = i & 127;
        float v = 0.0f;
        if (c < NCOL && k < DIN) v = W[k * NCOL + c];
        wT[c * WPITCH + k] = v;
    }

    // ---- GEMM phase: u_hat[j, 0..24] = x[j,:] @ W, via fp32 WMMA ----
    const int bcol  = lane & 15;             // B/C/D column (N index)
    const int arow  = (wid << 4) + (lane & 15);
    const int khalf = (lane >> 4) << 1;      // lanes 16-31 hold K+2,K+3

    for (int ch = 0; ch < NCHUNK; ++ch) {
        const float* srcf = xb + ch * CHUNK * DIN;
#if __has_builtin(__builtin_amdgcn_global_load_async_to_lds_b128)
        // CDNA5 async DMA: HBM -> LDS with no VGPR round-trip (ASYNCcnt-tracked)
        for (int i = tid; i < CHUNK * DIN / 4; i += 256) {
            int e   = i * 4;
            int row = e / DIN;
            int col = e - row * DIN;
            __builtin_amdgcn_global_load_async_to_lds_b128(
                (gv4i_p)(srcf + e),
                (lv4i_p)(&xch[row * XPITCH + col]),
                0, 0);
        }
        wait_async_zero();
#else
        // fallback: coalesced float4 stage through VGPRs
        const float4* src = (const float4*)srcf;
        for (int i = tid; i < CHUNK * DIN / 4; i += 256) {
            float4 v = src[i];
            int e   = i * 4;
            int row = e / DIN;
            int col = e - row * DIN;
            float* d = &xch[row * XPITCH + col];
            d[0] = v.x; d[1] = v.y; d[2] = v.z; d[3] = v.w;
        }
#endif
        __syncthreads();

        if (ch + 1 < NCHUNK) {   // pull next chunk toward L2 while we compute
            const char* nxt = (const char*)(xb + (ch + 1) * CHUNK * DIN);
            __builtin_prefetch(nxt + tid * 240, 0, 0);
        }

        v8f acc0 = {};
        v8f acc1 = {};
        #pragma unroll 6
        for (int s = 0; s < KSTEPS; ++s) {
            int k = (s << 2) + khalf;
            v2f a  = *(const v2f*)&xch[arow * XPITCH + k];
            v2f b0 = *(const v2f*)&wT[bcol * WPITCH + k];
            v2f b1 = *(const v2f*)&wT[(bcol + 16) * WPITCH + k];
            acc0 = __builtin_amdgcn_wmma_f32_16x16x4_f32(false, a, false, b0,
                                                         (short)0, acc0, false, false);
            acc1 = __builtin_amdgcn_wmma_f32_16x16x4_f32(false, a, false, b1,
                                                         (short)0, acc1, false, false);
        }

        // D layout: VGPR v -> M = v + 8*(lane>=16), N = lane%16.
        // Store all 32 columns unguarded (cols 25..31 are zero, never read).
        int rbase = ch * CHUNK + (wid << 4) + ((lane >> 4) << 3);
        #pragma unroll
        for (int v = 0; v < 8; ++v) {
            float* r = &uh[(rbase + v) * UPITCH];
            r[bcol]      = acc0[v];
            r[bcol + 16] = acc1[v];
        }
        __syncthreads();
    }

    // ---- routing phase: each thread owns j = tid and j = tid+256 ----
    float breg[2][NUM_CAP];
    #pragma unroll
    for (int jj = 0; jj < 2; ++jj)
        #pragma unroll
        for (int i = 0; i < NUM_CAP; ++i) breg[jj][i] = 0.0f;

    for (int it = 0; it < ROUTS; ++it) {
        float part[NCOL];
        #pragma unroll
        for (int p = 0; p < NCOL; ++p) part[p] = 0.0f;

        #pragma unroll
        for (int jj = 0; jj < 2; ++jj) {
            int j = tid + jj * 256;
            // softmax over the 5 capsules (thread-local)
            float m = breg[jj][0];
            #pragma unroll
            for (int i = 1; i < NUM_CAP; ++i) m = fmaxf(m, breg[jj][i]);
            float e[NUM_CAP]; float se = 0.0f;
            #pragma unroll
            for (int i = 0; i < NUM_CAP; ++i) { e[i] = __expf(breg[jj][i] - m); se += e[i]; }
            float inv = 1.0f / se;
            const float* u = &uh[j * UPITCH];
            #pragma unroll
            for (int i = 0; i < NUM_CAP; ++i) {
                float c = e[i] * inv;
                #pragma unroll
                for (int k = 0; k < DIM_CAP; ++k)
                    part[i * DIM_CAP + k] += c * u[i * DIM_CAP + k];
            }
        }

        // wave32 butterfly reduce, then 8-wave combine in LDS
        #pragma unroll
        for (int p = 0; p < NCOL; ++p)
            for (int off = 16; off; off >>= 1)
                part[p] += __shfl_xor(part[p], off, 32);
        if (lane == 0) {
            #pragma unroll
            for (int p = 0; p < NCOL; ++p) sred[wid * NCOL + p] = part[p];
        }
        __syncthreads();
        if (tid < NCOL) {
            float s = 0.0f;
            #pragma unroll
            for (int w = 0; w < 8; ++w) s += sred[w * NCOL + tid];
            sS[tid] = s;
        }
        __syncthreads();
        if (tid < NUM_CAP) {
            float nn = 0.0f;
            #pragma unroll
            for (int k = 0; k < DIM_CAP; ++k) { float v = sS[tid * DIM_CAP + k]; nn += v * v; }
            sInv[tid] = rsqrtf(nn + 1e-7f);
        }
        __syncthreads();

        if (it < ROUTS - 1) {
            #pragma unroll
            for (int jj = 0; jj < 2; ++jj) {
                int j = tid + jj * 256;
                const float* u = &uh[j * UPITCH];
                #pragma unroll
                for (int i = 0; i < NUM_CAP; ++i) {
                    float iv = sInv[i], acc = 0.0f;
                    #pragma unroll
                    for (int k = 0; k < DIM_CAP; ++k)
                        acc += sS[i * DIM_CAP + k] * iv * u[i * DIM_CAP + k];
                    breg[jj][i] = acc;
                }
            }
        }
        __syncthreads();
    }

    if (tid < NCOL)
        out[(size_t)b * NCOL + tid] = sS[tid] * sInv[tid / DIM_CAP];
}

extern "C" void kernel_launch(void* const* d_in, const int* in_sizes, int n_in,
                              void* d_out, int out_size, void* d_ws, size_t ws_size,
                              hipStream_t stream) {
    const float* x = (const float*)d_in[0];   // [1024,512,120] fp32
    const float* W = (const float*)d_in[1];   // [1,120,25] fp32
    float* out = (float*)d_out;               // [1024,5,5] fp32
    (void)in_sizes; (void)n_in; (void)out_size; (void)d_ws; (void)ws_size;

    size_t shmem = (size_t)SMEM_FLOATS * sizeof(float);  // ~148.9 KB (< 320 KB/WGP, 2 WG/WGP)
    caps_fused_kernel<<<1024, 256, shmem, stream>>>(x, W, out);
}